// CVAELoss_36275293782621
// MI455X (gfx1250) — compile-verified
//
#include <hip/hip_runtime.h>
#include <stdint.h>

#define B_N 4096
#define L_N 1024
#define F_N 5
#define LAT_N 128
#define PI_F 3.14159265358979323846f

// loss weights
#define W_POS 3.0f
#define W_TIME 0.5f
#define W_DIR 3.0f
#define W_EP 10.0f
#define W_LEN 2.0f
#define W_SPD 1.5f
#define W_DECEL 2.0f
#define W_DSM 2.5f
#define W_TSM 3.0f
#define W_KL 0.01f

typedef __attribute__((ext_vector_type(2))) float v2f;
typedef __attribute__((ext_vector_type(8))) float v8f;
typedef __attribute__((ext_vector_type(4))) int v4i;

typedef __attribute__((address_space(1))) v4i* gptr_v4i;
typedef __attribute__((address_space(3))) v4i* lptr_v4i;

// ---------------------------------------------------------------------------
// Wave32 sum-reduction on the matrix pipe: one V_WMMA_F32_16X16X4_F32 against
// an all-ones B.  A(16x4) layout: lanes 0-15 hold (K0,K1), lanes 16-31 hold
// (K2,K3) for rows M=lane%16.  With a=(v,0): rowsum(m) = v_m + v_{m+16},
// independent of B layout since B==1 everywhere.  D(16x16) layout: vgpr j,
// lanes 0-15 -> M=j, lanes 16-31 -> M=j+8, replicated across N.  Summing the
// 8 D components gives lanes 0-15: sum(rows 0..7), lanes 16-31: sum(rows
// 8..15); a final xor-16 shuffle yields the full 32-lane sum in every lane.
// Requires EXEC all-ones: only call from fully-converged code.
// ---------------------------------------------------------------------------
__device__ __forceinline__ float wave_sum_wmma(float v) {
  v2f a;
  a.x = v;
  a.y = 0.0f;
  v2f ones;
  ones.x = 1.0f;
  ones.y = 1.0f;
  v8f c = {};
  c = __builtin_amdgcn_wmma_f32_16x16x4_f32(false, a, false, ones, (short)0, c,
                                            false, false);
  float s = c[0] + c[1] + c[2] + c[3] + c[4] + c[5] + c[6] + c[7];
  s += __shfl_xor(s, 16, 32);
  return s;
}

#define N_TERMS 11
// term indices
// 0:S_pos 1:S_time 2:S_dir 3:S_spd 4:S_mask 5:S_decel 6:S_dcnt 7:S_dsm
// 8:S_tsm 9:S_acnt 10:S_kl

__global__ __launch_bounds__(256) void cvae_row_kernel(
    const float* __restrict__ recon, const float* __restrict__ target,
    const float* __restrict__ mu, const float* __restrict__ logvar,
    const float* __restrict__ plr, const int* __restrict__ lens,
    float* __restrict__ ws) {
  __shared__ __align__(16) float srow[L_N * F_N];  // 20 KB reconstruction row
  __shared__ float red[8][N_TERMS];

  const int b = blockIdx.x;
  const int tid = threadIdx.x;
  const int len = lens[b];

  const float* __restrict__ grow = recon + (size_t)b * (L_N * F_N);
  const float* __restrict__ trow = target + (size_t)b * (L_N * F_N);

  // ---- Stage reconstruction row into LDS (CDNA5 async path if available) ---
#if __has_builtin(__builtin_amdgcn_global_load_async_to_lds_b128)
  for (int i = tid; i < (L_N * F_N) / 4; i += 256) {
    __builtin_amdgcn_global_load_async_to_lds_b128(
        (gptr_v4i)(uintptr_t)(grow + i * 4),
        (lptr_v4i)(uint32_t)(uintptr_t)(srow + i * 4), 0, 0);
  }
#if __has_builtin(__builtin_amdgcn_s_wait_asynccnt)
  __builtin_amdgcn_s_wait_asynccnt(0);
#else
  asm volatile("s_wait_asynccnt 0" ::: "memory");
#endif
#else
  for (int i = tid; i < (L_N * F_N) / 4; i += 256) {
    ((float4*)srow)[i] = ((const float4*)grow)[i];
  }
#endif
  // Prefetch the target row while the barrier settles (global_prefetch_b8).
  for (int i = tid; i < L_N; i += 256) {
    __builtin_prefetch(trow + i * F_N, 0, 1);
  }
  __syncthreads();

  // ---- Per-thread partial sums over strided positions ----------------------
  float S[N_TERMS];
#pragma unroll
  for (int i = 0; i < N_TERMS; ++i) S[i] = 0.0f;

  for (int t = tid; t < L_N; t += 256) {
    const float* r = srow + t * F_N;
    const float r0 = r[0], r1 = r[1], r2 = r[2], r3 = r[3], r4 = r[4];
    const float* tg = trow + t * F_N;
    const float t0 = tg[0], t1 = tg[1], t2 = tg[2], t3 = tg[3], t4 = tg[4];

    if (t < len) {
      const float d0 = r0 - t0, d1 = r1 - t1, d2 = r2 - t2, d4 = r4 - t4;
      S[0] += d0 * d0 + d1 * d1;
      S[1] += d2 * d2;
      S[2] += 1.0f - cosf((r3 - t3) * PI_F);
      S[3] += d4 * d4;
      S[4] += 1.0f;
    }
    if (t < L_N - 1) {
      const float n3 = srow[(t + 1) * F_N + 3];
      const float n4 = srow[(t + 1) * F_N + 4];
      if (t < len - 1) {
        S[5] += fmaxf(n4 - r4, 0.0f);
        const float dd = (n3 - r3) * PI_F;
        const float w = atan2f(sinf(dd), cosf(dd));
        S[7] += w * w;
        S[6] += 1.0f;
      }
      if ((t < L_N - 2) && (t < len - 2)) {
        const float n0 = srow[(t + 1) * F_N + 0];
        const float n1 = srow[(t + 1) * F_N + 1];
        const float q0 = srow[(t + 2) * F_N + 0];
        const float q1 = srow[(t + 2) * F_N + 1];
        const float ax = q0 - 2.0f * n0 + r0;
        const float ay = q1 - 2.0f * n1 + r1;
        S[8] += sqrtf(ax * ax + ay * ay + 1e-8f);
        S[9] += 1.0f;
      }
    }
  }
  // KL partials: one latent element per thread (tid < 128)
  if (tid < LAT_N) {
    const float m = mu[(size_t)b * LAT_N + tid];
    float lv = logvar[(size_t)b * LAT_N + tid];
    lv = fminf(fmaxf(lv, -10.0f), 10.0f);
    S[10] = -0.5f * (1.0f + lv - m * m - expf(lv));
  }

  // ---- Intra-wave reduction via WMMA (EXEC all-ones here), then cross-wave -
  const int wave = tid >> 5;
  const int lane = tid & 31;
#pragma unroll
  for (int i = 0; i < N_TERMS; ++i) S[i] = wave_sum_wmma(S[i]);
  if (lane == 0) {
#pragma unroll
    for (int i = 0; i < N_TERMS; ++i) red[wave][i] = S[i];
  }
  __syncthreads();

  if (tid == 0) {
    float T[N_TERMS];
#pragma unroll
    for (int i = 0; i < N_TERMS; ++i) {
      float acc = 0.0f;
#pragma unroll
      for (int w = 0; w < 8; ++w) acc += red[w][i];
      T[i] = acc;
    }
    const float dcnt = fmaxf(T[6], 1.0f);
    const float acnt = fmaxf(T[9], 1.0f);
    const bool gt2 = len > 2;
    const int last = (len - 1 > 0) ? (len - 1) : 0;

    const float s_first = srow[0 * F_N + 4];
    const float s_last = srow[last * F_N + 4];
    const float start_pen = fmaxf(0.3f - s_first, 0.0f);
    const float end_pen = fmaxf(s_last - 0.2f, 0.0f);
    const float decel_term =
        gt2 ? (T[5] / dcnt + 0.5f * (start_pen + end_pen)) : 0.0f;
    const float dsm_term = gt2 ? (T[7] / dcnt) : 0.0f;
    const float tsm_term = gt2 ? (T[8] / acnt) : 0.0f;

    float ep = 0.0f;
    if (len > 0) {
      const float ex = srow[last * F_N + 0] - trow[last * F_N + 0];
      const float ey = srow[last * F_N + 1] - trow[last * F_N + 1];
      ep = 0.5f * (ex * ex + ey * ey);
    }
    const float ratio = (float)len / (float)L_N;
    const float dl = ratio - plr[b];
    const float len_term = dl * dl;

    const float other = W_EP * ep + W_LEN * len_term + W_DECEL * decel_term +
                        W_DSM * dsm_term + W_TSM * tsm_term + W_KL * T[10];

    ws[0 * B_N + b] = T[0];  // position sq sum
    ws[1 * B_N + b] = T[1];  // time sq sum
    ws[2 * B_N + b] = T[2];  // direction sum
    ws[3 * B_N + b] = T[3];  // speed sq sum
    ws[4 * B_N + b] = T[4];  // mask count
    ws[5 * B_N + b] = other; // per-row finalized terms (pre-weighted)
  }
}

__global__ __launch_bounds__(256) void cvae_final_kernel(
    const float* __restrict__ ws, float* __restrict__ out) {
  __shared__ float red[8][6];
  const int tid = threadIdx.x;
  float acc[6] = {0.f, 0.f, 0.f, 0.f, 0.f, 0.f};
  for (int i = tid; i < B_N; i += 256) {
#pragma unroll
    for (int k = 0; k < 6; ++k) acc[k] += ws[k * B_N + i];
  }
  const int wave = tid >> 5;
  const int lane = tid & 31;
#pragma unroll
  for (int k = 0; k < 6; ++k) acc[k] = wave_sum_wmma(acc[k]);
  if (lane == 0) {
#pragma unroll
    for (int k = 0; k < 6; ++k) red[wave][k] = acc[k];
  }
  __syncthreads();
  if (tid == 0) {
    float T[6];
#pragma unroll
    for (int k = 0; k < 6; ++k) {
      float a = 0.0f;
#pragma unroll
      for (int w = 0; w < 8; ++w) a += red[w][k];
      T[k] = a;
    }
    const float msum = T[4];
    const float loss = W_POS * T[0] / (2.0f * msum + 1e-8f) +
                       W_TIME * T[1] / (msum + 1e-8f) +
                       W_DIR * T[2] / (msum + 1e-8f) +
                       W_SPD * T[3] / (msum + 1e-8f) + T[5] / (float)B_N;
    out[0] = loss;
  }
}

extern "C" void kernel_launch(void* const* d_in, const int* in_sizes, int n_in,
                              void* d_out, int out_size, void* d_ws,
                              size_t ws_size, hipStream_t stream) {
  const float* recon = (const float*)d_in[0];
  const float* target = (const float*)d_in[1];
  const float* mu = (const float*)d_in[2];
  const float* logvar = (const float*)d_in[3];
  const float* plr = (const float*)d_in[4];
  const int* lens = (const int*)d_in[5];
  float* ws = (float*)d_ws;  // uses 6 * 4096 floats = 96 KB
  float* out = (float*)d_out;

  cvae_row_kernel<<<B_N, 256, 0, stream>>>(recon, target, mu, logvar, plr,
                                           lens, ws);
  cvae_final_kernel<<<1, 256, 0, stream>>>(ws, out);
}